// YoloLoss_77721728188525
// MI455X (gfx1250) — compile-verified
//
#include <hip/hip_runtime.h>

// Problem constants (from reference)
#define Bd 16
#define Wd 64
#define Hd 64
#define Ad 9
#define Cd 80
#define Nd 32
#define Gd (Bd*Wd*Hd*Ad)          // 589824 grid cells*anchors
#define NCELL (Bd*Wd*Hd)          // 65536 winner-table entries
#define NDW (Gd*5)                // 2949120 dwords in pred_object
#define NF4 (NDW/4)               // 737280 float4 loads (exact)

typedef float v2f __attribute__((ext_vector_type(2)));
typedef float v8f __attribute__((ext_vector_type(8)));

// ---------------------------------------------------------------------------
// 0) init: winner table = -1, accumulators = 0 (re-run every call: graph
//    replay does not re-poison workspace)
// ---------------------------------------------------------------------------
__global__ void yolo_init(int* __restrict__ winner, float* __restrict__ acc) {
  int t = blockIdx.x * blockDim.x + threadIdx.x;
  if (t < NCELL) winner[t] = -1;
  if (t < 8)     acc[t] = 0.0f;
}

// ---------------------------------------------------------------------------
// 1) scatter "last writer wins": winner[cell] = max n writing that cell
// ---------------------------------------------------------------------------
__global__ void yolo_winner(const float* __restrict__ tobj, int* __restrict__ winner) {
  int t = blockIdx.x * blockDim.x + threadIdx.x;
  if (t >= Bd * Nd) return;
  int b = t / Nd, n = t % Nd;
  float tox = tobj[t * 4 + 0] * 0.125f;   // true_object / DS
  float toy = tobj[t * 4 + 1] * 0.125f;
  int i = (int)tox; i = i < 0 ? 0 : (i > Wd - 1 ? Wd - 1 : i);
  int j = (int)toy; j = j < 0 ? 0 : (j > Hd - 1 ? Hd - 1 : j);
  atomicMax(&winner[(b * Wd + i) * Hd + j], n);
}

// ---------------------------------------------------------------------------
// 2) dense Sum(conf^2): stream pred_object as coalesced float4 (B128).
//    Grid exactly covers the array (2880*256*4 dwords == Gd*5), so there are
//    no bounds checks and no EXEC divergence. dword d is a conf value iff
//    d % 5 == 4; per thread m = (4t) % 5 selects component c when m == 4-c
//    (branchless cndmask). Selected squares are reduced through the matrix
//    pipe: D = ones(16x4) x B(4x16) + C accumulates exact-fp32 column sums;
//    each column sum appears twice across lanes in c[0] => wave partial =
//    Sum_lanes(c[0]) / 2.
// ---------------------------------------------------------------------------
__global__ void __launch_bounds__(256) yolo_conf_sq(const float4* __restrict__ pred4,
                                                    float* __restrict__ acc) {
  const int tid = threadIdx.x;
  const unsigned t = blockIdx.x * 256u + tid;          // 0 .. NF4-1 exactly

  float4 x = pred4[t];
  unsigned m = (4u * t) % 5u;
  float s0 = (m == 4u) ? x.x * x.x : 0.0f;
  float s1 = (m == 3u) ? x.y * x.y : 0.0f;
  float s2 = (m == 2u) ? x.z * x.z : 0.0f;
  float s3 = (m == 1u) ? x.w * x.w : 0.0f;

  v2f ones = {1.0f, 1.0f};
  v2f bm   = {s0 + s1, s2 + s3};
  v8f c    = {};
  // 8 args: (neg_a, A, neg_b, B, c_mod, C, reuse_a, reuse_b)
  c = __builtin_amdgcn_wmma_f32_16x16x4_f32(false, ones, false, bm,
                                            (short)0, c, false, false);

  __shared__ float red[256];
  red[tid] = c[0];
  __syncthreads();
  for (int s = 128; s > 0; s >>= 1) {
    if (tid < s) red[tid] += red[tid + s];
    __syncthreads();
  }
  if (tid == 0) atomicAdd(&acc[0], red[0] * 0.5f);     // /2: duplicated rows
}

// ---------------------------------------------------------------------------
// 3) sparse loss terms: one thread per (b,n) ground truth; only the cell
//    winner contributes (emulating .at[flat].set overwrite semantics).
// ---------------------------------------------------------------------------
__global__ void yolo_sparse(const float* __restrict__ cls,
                            const float* __restrict__ pred,
                            const float* __restrict__ tl,
                            const float* __restrict__ tobj,
                            const float* __restrict__ anch,
                            const int*   __restrict__ winner,
                            float*       __restrict__ acc) {
  int t = blockIdx.x * blockDim.x + threadIdx.x;
  if (t >= Bd * Nd) return;
  int b = t / Nd, n = t % Nd;

  float tox = tobj[t * 4 + 0] * 0.125f;
  float toy = tobj[t * 4 + 1] * 0.125f;
  float gtw = tobj[t * 4 + 2] * 0.125f;
  float gth = tobj[t * 4 + 3] * 0.125f;
  int i = (int)tox; i = i < 0 ? 0 : (i > Wd - 1 ? Wd - 1 : i);
  int j = (int)toy; j = j < 0 ? 0 : (j > Hd - 1 ? Hd - 1 : j);

  if (winner[(b * Wd + i) * Hd + j] != n) return;  // overwritten by later GT

  // anchor/gt center-IOU, best anchor (first-max wins, like jnp.argmax)
  float agt[Ad];
  int best = 0; float bestv = -1.0f;
  for (int a = 0; a < Ad; ++a) {
    float aw = anch[a * 2], ah = anch[a * 2 + 1];
    float inter = fminf(aw, gtw) * fminf(ah, gth);
    float uni   = aw * ah + gtw * gth - inter;
    float v = inter / uni;
    agt[a] = v;
    if (v > bestv) { bestv = v; best = a; }
  }

  const float* pc = pred + (unsigned)(((b * Wd + i) * Hd + j) * Ad) * 5u;

  // cells removed from noobj loss: obj_mask != 0  <=>  is_best || agt>thr
  float s1 = 0.0f;
  for (int a = 0; a < Ad; ++a) {
    if (a == best || agt[a] > 0.6f) { float cf = pc[a * 5 + 4]; s1 += cf * cf; }
  }

  // obj loss at best anchor: (conf - iou(pred_wh, gt_wh))^2
  float pw = pc[best * 5 + 2], ph = pc[best * 5 + 3];
  float inter = fminf(pw, gtw) * fminf(ph, gth);
  float uni   = pw * ph + gtw * gth - inter;
  float pg    = inter / uni;
  float cf    = pc[best * 5 + 4];
  float d2    = cf - pg;

  // xy loss (mirror reference fp order: (p - fm) - (t - fm))
  float px = pc[best * 5 + 0], py = pc[best * 5 + 1];
  float fx = (float)i, fy = (float)j;
  float dx = (px - fx) - (tox - fx);
  float dy = (py - fy) - (toy - fy);

  // wh loss with scale weight sw = 2 - (w/OUT)*(h/OUT)
  float sw = 2.0f - (gtw * (1.0f / 64.0f)) * (gth * (1.0f / 64.0f));
  float dw = pw - gtw, dh = ph - gth;

  // class score loss over C=80 at the best-anchor cell
  const float* cs  = cls + (unsigned)(((b * Wd + i) * Hd + j) * Ad + best) * (unsigned)Cd;
  const float* tlp = tl + (unsigned)t * (unsigned)Cd;
  float s5 = 0.0f;
  for (int c2 = 0; c2 < Cd; ++c2) { float d = cs[c2] - tlp[c2]; s5 += d * d; }

  atomicAdd(&acc[1], s1);
  atomicAdd(&acc[2], d2 * d2);
  atomicAdd(&acc[3], dx * dx + dy * dy);
  atomicAdd(&acc[4], sw * (dw * dw + dh * dh));
  atomicAdd(&acc[5], s5);
}

// ---------------------------------------------------------------------------
// 4) finalize the six scalar outputs
// ---------------------------------------------------------------------------
__global__ void yolo_final(const float* __restrict__ acc, float* __restrict__ out) {
  const float invG = 1.0f / (float)Gd;
  float noobj = 1.0f * 0.5f * (acc[0] - acc[1]) * invG;           // SCALE_NOOBJ
  float obj   = 5.0f * 0.5f * acc[2] * invG;                      // SCALE_OBJ
  float xy    = 5.0f * 0.5f * acc[3] * invG * 0.5f;               // mean over 2G
  float wh    = 5.0f * 0.5f * acc[4] * invG * 0.5f;
  float score = 5.0f * 0.5f * acc[5] * invG * (1.0f / (float)Cd); // mean over C*G
  float total = (noobj + obj + xy + wh + score) * 0.25f;
  out[0] = total;
  out[1] = noobj * 0.25f;
  out[2] = obj   * 0.25f;
  out[3] = score * 0.25f;
  out[4] = xy    * 0.25f;
  out[5] = wh    * 0.25f;
}

// ---------------------------------------------------------------------------
extern "C" void kernel_launch(void* const* d_in, const int* in_sizes, int n_in,
                              void* d_out, int out_size, void* d_ws, size_t ws_size,
                              hipStream_t stream) {
  (void)in_sizes; (void)n_in; (void)out_size; (void)ws_size;
  // inputs: [0]=epoch(int,unused) [1]=cls_score [2]=pred_object [3]=true_label
  //         [4]=true_object [5]=anchor_box [6]=fm_cord(unused: cancels exactly)
  const float* cls  = (const float*)d_in[1];
  const float* pred = (const float*)d_in[2];
  const float* tl   = (const float*)d_in[3];
  const float* tobj = (const float*)d_in[4];
  const float* anch = (const float*)d_in[5];

  int*   winner = (int*)d_ws;                                          // 65536 ints
  float* acc    = (float*)((char*)d_ws + (size_t)NCELL * sizeof(int)); // 8 floats
  float* out    = (float*)d_out;

  yolo_init   <<<(NCELL + 255) / 256, 256, 0, stream>>>(winner, acc);
  yolo_winner <<<(Bd * Nd + 255) / 256, 256, 0, stream>>>(tobj, winner);
  yolo_conf_sq<<<NF4 / 256, 256, 0, stream>>>((const float4*)pred, acc); // 2880 blocks
  yolo_sparse <<<(Bd * Nd + 255) / 256, 256, 0, stream>>>(cls, pred, tl, tobj, anch, winner, acc);
  yolo_final  <<<1, 1, 0, stream>>>(acc, out);
}